// Unfold_45174466019399
// MI455X (gfx1250) — compile-verified
//
#include <hip/hip_runtime.h>

// 3D unfold with KERNEL==STRIDE==(4,4,4): pure permutation, HBM-bound.
// x: (8,16,64,64,64) f32  ->  out: (8,16,4,4,4,4096) f32
// out[b,c,i,j,k, h*256+w*16+d] = x[b,c, 4h+i, 4w+j, 4d+k]
//
// Each thread: 64B contiguous load (D-range [16q,16q+16)) as 4x float4,
// 4x4 register transpose, 4x float4 coalesced stores (one per k-plane).
// Non-temporal hints on both streams (no reuse; 256MB streams > 192MB L2).

typedef __attribute__((ext_vector_type(4))) float v4f;

__global__ __launch_bounds__(256)
void unfold443_permute_kernel(const float* __restrict__ in,
                              float* __restrict__ out) {
    // t decomposition (q fastest -> contiguous loads AND stores across a wave):
    // t = ((((bc*4 + i)*4 + j)*16 + h)*16 + w)*4 + q
    const unsigned t  = blockIdx.x * 256u + threadIdx.x;
    const unsigned q  =  t         & 3u;   // d-quad: d = 4q + dd
    const unsigned w  = (t >> 2)   & 15u;
    const unsigned h  = (t >> 6)   & 15u;
    const unsigned j  = (t >> 10)  & 3u;
    const unsigned i  = (t >> 12)  & 3u;
    const unsigned bc =  t >> 14;          // b*16 + c, 0..127

    // Input: (((bc*64 + 4h+i)*64 + 4w+j)*64 + 16q) floats; 64B aligned.
    const size_t inoff = (size_t)bc * 262144u
                       + (size_t)(4u * h + i) * 4096u
                       + (size_t)(4u * w + j) * 64u
                       + (size_t)q * 16u;
    const v4f* __restrict__ ip = (const v4f*)(in + inoff);
    // r[dd][k] = x at D = 16q + 4*dd + k
    const v4f r0 = __builtin_nontemporal_load(ip + 0);
    const v4f r1 = __builtin_nontemporal_load(ip + 1);
    const v4f r2 = __builtin_nontemporal_load(ip + 2);
    const v4f r3 = __builtin_nontemporal_load(ip + 3);

    // Output base: (bc*64 + i*16 + j*4 + k)*4096 + h*256 + w*16 + 4q
    const size_t obase = (size_t)(bc * 64u + i * 16u + j * 4u) * 4096u
                       + (size_t)h * 256u + (size_t)w * 16u + (size_t)q * 4u;

#pragma unroll
    for (int k = 0; k < 4; ++k) {
        // o[dd] = element for d = 4q + dd in k-plane -> 16B contiguous store
        v4f o;
        o.x = r0[k];
        o.y = r1[k];
        o.z = r2[k];
        o.w = r3[k];
        __builtin_nontemporal_store(o, (v4f*)(out + obase + (size_t)k * 4096u));
    }
}

extern "C" void kernel_launch(void* const* d_in, const int* in_sizes, int n_in,
                              void* d_out, int out_size, void* d_ws, size_t ws_size,
                              hipStream_t stream) {
    (void)in_sizes; (void)n_in; (void)out_size; (void)d_ws; (void)ws_size;
    const float* x  = (const float*)d_in[0];
    float* out      = (float*)d_out;

    // 8*16 (bc) * 4*4 (i,j) * 16*16 (h,w) * 4 (q) = 2,097,152 threads,
    // 16 elements each = 33,554,432 elements total.
    const unsigned total_threads = 2097152u;
    dim3 block(256u);
    dim3 grid(total_threads / 256u);   // 8192 blocks
    hipLaunchKernelGGL(unfold443_permute_kernel, grid, block, 0, stream, x, out);
}